// DAS_59820304499216
// MI455X (gfx1250) — compile-verified
//
#include <hip/hip_runtime.h>
#include <math.h>

// ---------------- problem constants ----------------
#define NBATCH 8
#define NSENS  128
#define NT     4096
#define NXDIM  256
#define NYDIM  256
#define NPIX   (NXDIM * NYDIM)          // 65536

// ---------------- tiling constants -----------------
#define TLDS    2368                     // staged samples per row (max t = 2326, multiple of 16 f32)
#define SCH     2                        // sensors per chunk
#define ROWS    (NBATCH * SCH)           // 16 LDS rows per buffer
#define NCH     (NSENS / SCH)            // 64 chunks
#define THREADS 256
#define PXB     512                      // pixels per block
#define PXT     (PXB / THREADS)          // 2 pixels per thread

static_assert(2 * ROWS * TLDS * 4 <= 320 * 1024, "double-buffered stage must fit 320KB LDS");

typedef unsigned int u32x4 __attribute__((ext_vector_type(4)));
typedef unsigned int u32x8 __attribute__((ext_vector_type(8)));

// Issue one Tensor-Data-Mover DMA for a chunk: a 3-D tile
//   X = TLDS contiguous samples, Y = SCH sensors (stride NT), Z = NBATCH batches (stride NSENS*NT)
// written contiguously into LDS at buf*ROWS*TLDS*4, i.e. row (b*SCH+cs) at the layout
// the gather loop expects. D# bitfields per CDNA5 ISA ch.8 (groups 0-3).
__device__ __forceinline__ void issue_chunk_tdm(const float* __restrict__ data, int s0, int buf) {
    const unsigned long long ga =
        (unsigned long long)(uintptr_t)data + (unsigned long long)s0 * (NT * 4ull);

    u32x4 g0;
    g0.x = 1u;                                                   // count=1, user descriptor
    g0.y = (unsigned)((buf * ROWS * TLDS) * 4);                  // lds_addr (bytes)
    g0.z = (unsigned)(ga & 0xFFFFFFFFull);                       // global_addr[31:0]
    g0.w = (unsigned)((ga >> 32) & 0x01FFFFFFull) | (2u << 30);  // global_addr[56:32] | type=2

    u32x8 g1;
    g1[0] = 0x00020000u;                                         // data_size = 2 (4 bytes)
    g1[1] = ((unsigned)NT & 0xFFFFu) << 16;                      // tensor_dim0[15:0] = 4096
    g1[2] = (((unsigned)NT >> 16) & 0xFFFFu)                     // tensor_dim0[31:16]
          | (((unsigned)NSENS & 0xFFFFu) << 16);                 // tensor_dim1[15:0] = 128
    g1[3] = ((unsigned)TLDS << 16);                              // tensor_dim1[31:16]=0 | tile_dim0 = 2368
    g1[4] = (unsigned)SCH | ((unsigned)NBATCH << 16);            // tile_dim1 = 2 | tile_dim2 = 8
    g1[5] = (unsigned)NT;                                        // tensor_dim0_stride[31:0] = 4096
    g1[6] = (((unsigned)(NSENS * NT) & 0xFFFFu) << 16);          // stride0[47:32]=0 | stride1[15:0]
    g1[7] = ((unsigned)(NSENS * NT) >> 16);                      // tensor_dim1_stride[47:16] = 8

    u32x4 g2 = {(unsigned)NBATCH, 1u, 0u, 0u};                   // tensor_dim2=8, tensor_dim3=1
    u32x4 g3 = {0u, 0u, 0u, 0u};

    asm volatile("tensor_load_to_lds %0, %1, %2, %3"
                 :
                 : "s"(g0), "s"(g1), "s"(g2), "s"(g3)
                 : "memory");
}

extern "C" __global__ __launch_bounds__(THREADS)
void das_main(const float* __restrict__ data,   // (B, S, T) f32
              const int*   __restrict__ sxy,    // (S, 2) int32
              float*       __restrict__ img) {  // (B, P) raw sums
    extern __shared__ float lds[];              // 2 * ROWS * TLDS floats (303,104 B)

    const int tid = threadIdx.x;
    const int p0  = blockIdx.x * PXB + tid;

    float fx[PXT], fy[PXT];
    float acc[PXT][NBATCH];
#pragma unroll
    for (int j = 0; j < PXT; ++j) {
        const int p = p0 + j * THREADS;
        fx[j] = (float)(p >> 8);     // pixel row (ix)
        fy[j] = (float)(p & 255);    // pixel col (iy)
#pragma unroll
        for (int b = 0; b < NBATCH; ++b) acc[j][b] = 0.0f;
    }

    // prologue: wave 0 kicks off the DMA of chunk 0 into buffer 0
    if (tid < 32) issue_chunk_tdm(data, 0, 0);

    for (int c = 0; c < NCH; ++c) {
        const int buf = c & 1;

        if (c + 1 < NCH) {
            // overlap: start next chunk's DMA into the other buffer, then retire
            // the previous DMA (tensor ops complete in order per wave).
            if (tid < 32) issue_chunk_tdm(data, (c + 1) * SCH, buf ^ 1);
            __builtin_amdgcn_s_wait_tensorcnt(1);
        } else {
            __builtin_amdgcn_s_wait_tensorcnt(0);
        }
        __syncthreads();   // publish wave-0's completed DMA to all 8 waves

        const int s0 = c * SCH;
#pragma unroll
        for (int cs = 0; cs < SCH; ++cs) {
            const int   s  = s0 + cs;
            const float sx = (float)sxy[2 * s + 0];
            const float sy = (float)sxy[2 * s + 1];
            const float* rowbase = &lds[(buf * ROWS + cs) * TLDS]; // batch stride = SCH*TLDS
#pragma unroll
            for (int j = 0; j < PXT; ++j) {
                const float dx  = (sx - fx[j]) * 1.0e-3f;
                const float dy  = (sy - fy[j]) * 1.0e-3f;
                const float dis = sqrtf(dx * dx + dy * dy);
                const int   t   = (int)(dis / 1550.0f / 1.0e-7f);  // same op order as reference
#pragma unroll
                for (int b = 0; b < NBATCH; ++b)
                    acc[j][b] += rowbase[b * (SCH * TLDS) + t];    // ds_load_b32 gather
            }
        }
        __syncthreads();   // buffer `buf` is re-filled by the DMA issued at iteration c+1
    }

#pragma unroll
    for (int j = 0; j < PXT; ++j) {
        const int p = p0 + j * THREADS;
#pragma unroll
        for (int b = 0; b < NBATCH; ++b)
            img[b * NPIX + p] = acc[j][b];
    }
}

extern "C" __global__ __launch_bounds__(256)
void das_minmax(const float* __restrict__ img, float* __restrict__ mm) {
    __shared__ float smn[256];
    __shared__ float smx[256];
    const int b = blockIdx.x;
    const float* p = img + b * NPIX;
    float mn = 3.402823466e38f, mx = -3.402823466e38f;
    for (int i = threadIdx.x; i < NPIX; i += 256) {
        const float v = p[i];
        mn = fminf(mn, v);
        mx = fmaxf(mx, v);
    }
    smn[threadIdx.x] = mn;
    smx[threadIdx.x] = mx;
    __syncthreads();
    for (int o = 128; o > 0; o >>= 1) {
        if (threadIdx.x < o) {
            smn[threadIdx.x] = fminf(smn[threadIdx.x], smn[threadIdx.x + o]);
            smx[threadIdx.x] = fmaxf(smx[threadIdx.x], smx[threadIdx.x + o]);
        }
        __syncthreads();
    }
    if (threadIdx.x == 0) {
        mm[2 * b + 0] = smn[0];
        mm[2 * b + 1] = smx[0];
    }
}

extern "C" __global__ __launch_bounds__(256)
void das_norm(float* __restrict__ out, const float* __restrict__ mm) {
    const int i = blockIdx.x * 256 + threadIdx.x;   // over B*NPIX
    const int b = i >> 16;                          // NPIX == 65536
    const float mn = mm[2 * b + 0];
    const float mx = mm[2 * b + 1];
    out[i] = (out[i] - mn) / (mx - mn);
}

extern "C" void kernel_launch(void* const* d_in, const int* in_sizes, int n_in,
                              void* d_out, int out_size, void* d_ws, size_t ws_size,
                              hipStream_t stream) {
    const float* data = (const float*)d_in[0];   // (B,S,T) f32
    const int*   sxy  = (const int*)d_in[1];     // (S,2) int32
    float*       out  = (float*)d_out;           // (B,NX,NY) f32
    float*       mm   = (float*)d_ws;            // 2*B floats of scratch

    const size_t shbytes = (size_t)(2 * ROWS * TLDS) * sizeof(float);  // 303,104 B

    das_main  <<<NPIX / PXB,            THREADS, shbytes, stream>>>(data, sxy, out);
    das_minmax<<<NBATCH,                256,     0,       stream>>>(out, mm);
    das_norm  <<<(NBATCH * NPIX) / 256, 256,     0,       stream>>>(out, mm);
}